// Dynamic_Cnn_Wrapper_52295521796492
// MI455X (gfx1250) — compile-verified
//
#include <hip/hip_runtime.h>

// ---------------------------------------------------------------------------
// Dynamic CNN (per-position dynamic 5-tap causal temporal conv, 4 heads)
//   x: (128, 512, 14, 14) f32, W: (20, 512) f32, out: (128, 512, 14, 14) f32
// Kernel A: fp32 WMMA GEMM (logits) + softmax -> ws weights[t][b][20]
// Kernel B: bandwidth-bound 5-tap weighted combine
// ---------------------------------------------------------------------------

typedef float v2f __attribute__((ext_vector_type(2)));
typedef float v8f __attribute__((ext_vector_type(8)));

#define T_SEG 8
#define NSEG  16          // 128 / 8
#define CCH   512
#define HW    196         // 14*14
#define B_TOT (NSEG * HW) // 3136
#define HEADS 4
#define KTAPS 5
#define O20   (HEADS * KTAPS) // 20
#define RTILE 13          // ceil(196/16) row tiles of 16

// ---------------- Kernel A: logits GEMM (WMMA f32) + softmax ----------------
// grid.x = NSEG*T_SEG*RTILE = 1664, block = 32 (one wave; EXEC all ones)
__global__ __launch_bounds__(32)
void dyncnn_logits_softmax(const float* __restrict__ x,
                           const float* __restrict__ W,
                           float* __restrict__ wsm)
{
    const int lane = threadIdx.x;
    const int tile = blockIdx.x % RTILE;
    const int tn   = blockIdx.x / RTILE;
    const int t    = tn % T_SEG;
    const int ni   = tn / T_SEG;

    const int half = lane >> 4;       // 0: lanes 0-15, 1: lanes 16-31
    const int l15  = lane & 15;

    // ---- A fragment: A[M=hw][K=c] = x[(ni*8+t), c, hw]
    // fp32 16x4 A layout: lanes 0-15 hold K={0,1}, lanes 16-31 hold K={2,3}
    int hw  = tile * 16 + l15;
    int hwc = (hw < HW - 1) ? hw : (HW - 1);          // clamp (no divergence)
    const int cA = half ? 2 : 0;
    const float* aptr = x + ((size_t)((ni * T_SEG + t) * CCH) + cA) * HW + hwc;

    // ---- B fragment: B[K=c][N=o] = W[o][c]; float2 loads (c even -> aligned)
    const float2* __restrict__ W2 = (const float2*)W;
    int o0 = l15;                                     // N tile 0: o = 0..15
    int o1 = o0 + 16; if (o1 > O20 - 1) o1 = O20 - 1; // N tile 1: o = 16..19 (clamped)
    int bofs0 = o0 * (CCH / 2) + (cA >> 1);
    int bofs1 = o1 * (CCH / 2) + (cA >> 1);

    v8f acc0 = {}; v8f acc1 = {};
    for (int k4 = 0; k4 < CCH; k4 += 4) {
        v2f a;
        a.x = aptr[0];          // c = k4 + cA
        a.y = aptr[HW];         // c = k4 + cA + 1
        aptr += 4 * HW;
        float2 f0 = W2[bofs0];
        float2 f1 = W2[bofs1];
        bofs0 += 2; bofs1 += 2;
        v2f b0; b0.x = f0.x; b0.y = f0.y;
        v2f b1; b1.x = f1.x; b1.y = f1.y;
        // D = A(16x4) * B(4x16) + C, exact fp32 matrix math
        acc0 = __builtin_amdgcn_wmma_f32_16x16x4_f32(
                   false, a, false, b0, (short)0, acc0, false, false);
        acc1 = __builtin_amdgcn_wmma_f32_16x16x4_f32(
                   false, a, false, b1, (short)0, acc1, false, false);
    }

    // ---- spill C tile (16 rows x 20 valid cols) to LDS
    // C layout: lanes 0-15: VGPR r = C[M=r][N=lane]; lanes 16-31: C[M=8+r][N=lane-16]
    __shared__ float Lsm[16 * 32];
    {
        const int mbase = half * 8;
        #pragma unroll
        for (int r = 0; r < 8; ++r) {
            Lsm[(mbase + r) * 32 + l15]      = acc0[r];
            Lsm[(mbase + r) * 32 + 16 + l15] = acc1[r];
        }
    }
    __syncthreads();

    // ---- softmax over the 5 taps; 64 (row,head) tasks -> 2 per lane
    #pragma unroll
    for (int i = 0; i < 2; ++i) {
        const int task = lane * 2 + i;
        const int rr = task >> 2;     // row 0..15
        const int h  = task & 3;      // head 0..3
        const float* lp = &Lsm[rr * 32 + h * KTAPS];
        float v0 = lp[0], v1 = lp[1], v2 = lp[2], v3 = lp[3], v4 = lp[4];
        float m = fmaxf(fmaxf(fmaxf(v0, v1), fmaxf(v2, v3)), v4);
        float e0 = __expf(v0 - m), e1 = __expf(v1 - m), e2 = __expf(v2 - m),
              e3 = __expf(v3 - m), e4 = __expf(v4 - m);
        float inv = 1.0f / (e0 + e1 + e2 + e3 + e4);
        const int hwr = tile * 16 + rr;
        if (hwr < HW) {
            size_t base = (((size_t)t * B_TOT) + (size_t)ni * HW + hwr) * O20
                          + (size_t)h * KTAPS;
            wsm[base + 0] = e0 * inv;
            wsm[base + 1] = e1 * inv;
            wsm[base + 2] = e2 * inv;
            wsm[base + 3] = e3 * inv;
            wsm[base + 4] = e4 * inv;
        }
    }
}

// ---------------- Kernel B: causal 5-tap dynamic combine --------------------
// one thread per output element; 100352 elements per (nt) slab, 100352%256==0
// so t / ni are uniform per block -> causal branch is wave-uniform.
__global__ __launch_bounds__(256)
void dyncnn_combine(const float* __restrict__ x,
                    const float* __restrict__ wsm,
                    float* __restrict__ out)
{
    const size_t idx = (size_t)blockIdx.x * 256 + threadIdx.x;
    const int hw  = (int)(idx % HW);
    const int c   = (int)((idx / HW) % CCH);
    const int nti = (int)(idx / ((size_t)HW * CCH));
    const int ni  = nti >> 3;
    const int t   = nti & 7;
    const int h   = c >> 7;   // c / 128

    const float* wp = wsm + (((size_t)t * B_TOT) + (size_t)ni * HW + hw) * O20
                          + (size_t)h * KTAPS;
    const float* xb = x + (((size_t)(ni * T_SEG)) * CCH + c) * HW + hw;
    const size_t tstride = (size_t)CCH * HW;

    float acc = 0.0f;
    #pragma unroll
    for (int k = 0; k < KTAPS; ++k) {
        const int ts = t + k - 4;               // causal: zero-padded left
        if (ts >= 0)
            acc = fmaf(xb[(size_t)ts * tstride], wp[k], acc);
    }
    out[idx] = acc;
}

// ---------------------------------------------------------------------------
extern "C" void kernel_launch(void* const* d_in, const int* in_sizes, int n_in,
                              void* d_out, int out_size, void* d_ws, size_t ws_size,
                              hipStream_t stream) {
    (void)in_sizes; (void)n_in; (void)out_size; (void)ws_size;
    const float* x   = (const float*)d_in[0];   // (128,512,14,14) f32
    const float* W   = (const float*)d_in[1];   // (20,512) f32
    float*       out = (float*)d_out;           // (128,512,14,14) f32
    float*       wsm = (float*)d_ws;            // needs T*B*20*4 = ~2 MB scratch

    // Kernel A: 1664 single-wave workgroups
    dyncnn_logits_softmax<<<dim3(NSEG * T_SEG * RTILE), dim3(32), 0, stream>>>(x, W, wsm);

    // Kernel B: 12,845,056 elements / 256 = 50176 blocks (exact)
    const size_t total = (size_t)NSEG * T_SEG * CCH * HW;
    dyncnn_combine<<<dim3((unsigned)(total / 256)), dim3(256), 0, stream>>>(x, wsm, out);
}